// GaussianRenderer_38104949850253
// MI455X (gfx1250) — compile-verified
//
#include <hip/hip_runtime.h>
#include <hip/hip_bf16.h>

#define N_G   2048
#define IMG_H 128
#define IMG_W 128
#define REC   12            // floats per gaussian record
#define CHUNK 128           // gaussians staged to LDS per block iteration
#define NCHUNK (N_G / CHUNK)
#define CHUNK_FLTS (CHUNK * REC)   // 1536 floats = 6KB per buffer

typedef __attribute__((ext_vector_type(2))) float v2f;
typedef __attribute__((ext_vector_type(8))) float v8f;
typedef __attribute__((ext_vector_type(4))) unsigned int v4u;
typedef __attribute__((ext_vector_type(4))) int v4i;
typedef __attribute__((ext_vector_type(8))) int v8i;

// ---------------------------------------------------------------------------
// Kernel 1: per-gaussian preprocessing -> 12-float record + depth key
// record: [c0..c5, 0, 0, opac, r, g, b]  where  maha = sum_k f_k(pixel)*c_k
// ---------------------------------------------------------------------------
__global__ void gs_prep(const float* __restrict__ off, const float* __restrict__ scl,
                        const float* __restrict__ opa, const float* __restrict__ rot,
                        const float* __restrict__ col, const float* __restrict__ vm,
                        const float* __restrict__ pm, float* __restrict__ rec,
                        float* __restrict__ keys, int* __restrict__ idx) {
  int g = blockIdx.x * blockDim.x + threadIdx.x;
  if (g >= N_G) return;

  float qw = rot[g*4+0], qx = rot[g*4+1], qy = rot[g*4+2], qz = rot[g*4+3];
  float R[3][3] = {
    {1.f-2.f*(qy*qy+qz*qz), 2.f*(qx*qy-qw*qz),     2.f*(qx*qz+qw*qy)},
    {2.f*(qx*qy+qw*qz),     1.f-2.f*(qx*qx+qz*qz), 2.f*(qy*qz-qw*qx)},
    {2.f*(qx*qz-qw*qy),     2.f*(qy*qz+qw*qx),     1.f-2.f*(qx*qx+qy*qy)}};
  float s[3] = {scl[g*3+0], scl[g*3+1], scl[g*3+2]};
  float RS[3][3];
  #pragma unroll
  for (int i = 0; i < 3; ++i)
    #pragma unroll
    for (int j = 0; j < 3; ++j) RS[i][j] = R[i][j] * s[j];
  float S3[3][3];
  #pragma unroll
  for (int i = 0; i < 3; ++i)
    #pragma unroll
    for (int j = 0; j < 3; ++j)
      S3[i][j] = RS[i][0]*RS[j][0] + RS[i][1]*RS[j][1] + RS[i][2]*RS[j][2];

  float o0 = off[g*3+0], o1 = off[g*3+1], o2 = off[g*3+2];
  float mc[4];
  #pragma unroll
  for (int r = 0; r < 4; ++r)
    mc[r] = vm[r*4+0]*o0 + vm[r*4+1]*o1 + vm[r*4+2]*o2 + vm[r*4+3];
  float depth = fmaxf(mc[2], 0.01f);

  float nd[4];
  #pragma unroll
  for (int r = 0; r < 4; ++r)
    nd[r] = pm[r*4+0]*mc[0] + pm[r*4+1]*mc[1] + pm[r*4+2]*mc[2] + pm[r*4+3]*mc[3];
  float invw = 1.0f / nd[3];
  float mx = (nd[0]*invw + 1.0f) * 0.5f * (float)IMG_W;
  float my = (1.0f - nd[1]*invw) * 0.5f * (float)IMG_H;

  float fxp = pm[0]  * (float)IMG_W * 0.5f;
  float fyp = pm[5]  * (float)IMG_H * 0.5f;
  float tz = depth, itz = 1.0f / tz;
  float J[2][3] = {
    {fxp*itz, 0.0f,   -fxp*mc[0]*itz*itz},
    {0.0f,    fyp*itz, -fyp*mc[1]*itz*itz}};
  float JW[2][3];
  #pragma unroll
  for (int a = 0; a < 2; ++a)
    #pragma unroll
    for (int c = 0; c < 3; ++c)
      JW[a][c] = J[a][0]*vm[0*4+c] + J[a][1]*vm[1*4+c] + J[a][2]*vm[2*4+c];

  float C2[2][2] = {{0.f,0.f},{0.f,0.f}};
  #pragma unroll
  for (int a = 0; a < 2; ++a)
    #pragma unroll
    for (int b = 0; b < 2; ++b)
      #pragma unroll
      for (int i = 0; i < 3; ++i)
        #pragma unroll
        for (int j = 0; j < 3; ++j)
          C2[a][b] += JW[a][i] * S3[i][j] * JW[b][j];

  float A  = C2[0][0] + 0.3f;
  float Bc = C2[0][1];
  float Cc = C2[1][1] + 0.3f;
  float det = fmaxf(A*Cc - Bc*Bc, 1e-6f);
  float idet = 1.0f / det;
  float ia = Cc*idet, ib = -Bc*idet, ic = A*idet;

  float op = 1.0f / (1.0f + __expf(-opa[g]));

  float* r12 = rec + g*REC;
  r12[0] = ia;
  r12[1] = 2.0f*ib;
  r12[2] = ic;
  r12[3] = -2.0f*(ia*mx + ib*my);
  r12[4] = -2.0f*(ib*mx + ic*my);
  r12[5] = ia*mx*mx + 2.0f*ib*mx*my + ic*my*my;
  r12[6] = 0.0f;
  r12[7] = 0.0f;
  r12[8] = op;
  r12[9]  = col[g*3+0];
  r12[10] = col[g*3+1];
  r12[11] = col[g*3+2];
  keys[g] = depth;
  idx[g]  = g;
}

// ---------------------------------------------------------------------------
// Kernel 2: single-workgroup bitonic key/index sort (ascending depth)
// ---------------------------------------------------------------------------
__global__ void gs_sort(float* __restrict__ keys, int* __restrict__ idx) {
  __shared__ float sk[N_G];
  __shared__ int   si[N_G];
  int tid = threadIdx.x;
  for (int i = tid; i < N_G; i += blockDim.x) { sk[i] = keys[i]; si[i] = idx[i]; }
  __syncthreads();
  for (int k = 2; k <= N_G; k <<= 1) {
    for (int j = k >> 1; j > 0; j >>= 1) {
      for (int i = tid; i < N_G; i += blockDim.x) {
        int ixj = i ^ j;
        if (ixj > i) {
          bool up = ((i & k) == 0);
          float a = sk[i], b = sk[ixj];
          if ((a > b) == up) {
            sk[i] = b; sk[ixj] = a;
            int t = si[i]; si[i] = si[ixj]; si[ixj] = t;
          }
        }
      }
      __syncthreads();
    }
  }
  for (int i = tid; i < N_G; i += blockDim.x) { keys[i] = sk[i]; idx[i] = si[i]; }
}

// ---------------------------------------------------------------------------
// Kernel 3: gather records into depth order (contiguous stream for render)
// ---------------------------------------------------------------------------
__global__ void gs_gather(const float* __restrict__ rec, const int* __restrict__ idx,
                          float* __restrict__ recS) {
  int i = blockIdx.x * blockDim.x + threadIdx.x;
  if (i >= N_G) return;
  int s = idx[i];
  const float4* src = (const float4*)(rec + s*REC);
  float4* dst = (float4*)(recS + i*REC);
  dst[0] = src[0]; dst[1] = src[1]; dst[2] = src[2];
}

// ---------------------------------------------------------------------------
// TDM: DMA one 6KB chunk (1536 contiguous f32, a 1-D tile) global -> LDS.
// D# built per CDNA5 ISA 8.3/8.4:
//   group0: count=1 | lds_addr | global_addr[56:0] | type=2
//   group1: data_size=4B, tensor_dim0=tile_dim0=1536, dim0_stride=1536
//   remaining groups: zero (tile is 1-D)
// Tracked by TENSORcnt; issuing wave retires with s_wait_tensorcnt.
// This toolchain exposes the 6-arg builtin:
//   (uint32x4, int32x8, int32x4, int32x4, int32x8, i32 cpol)
// ---------------------------------------------------------------------------
__device__ __forceinline__ void tdm_load_chunk(const float* __restrict__ gsrc,
                                               unsigned lds_byte_addr) {
  unsigned long long ga = (unsigned long long)(uintptr_t)gsrc;
  v4u g0;
  g0.x = 1u;                                     // count=1 (user mode)
  g0.y = lds_byte_addr;                          // D#.lds_addr
  g0.z = (unsigned)ga;                           // D#.global_addr[31:0]
  g0.w = (unsigned)((ga >> 32) & 0x01FFFFFFull)  // D#.global_addr[56:32]
       | (2u << 30);                             // D#.type = 2 ("image")
  v8i g1;
  g1[0] = 0x00020000;                            // data_size = 2 -> 4 bytes
  g1[1] = (int)((CHUNK_FLTS & 0xFFFF) << 16);    // tensor_dim0 lo16 @ bits[63:48]
  g1[2] = 0;                                     // tensor_dim0 hi / tensor_dim1 lo
  g1[3] = (int)(CHUNK_FLTS << 16);               // tile_dim0 @ bits[127:112]
  g1[4] = 0;                                     // tile_dim1=0 (1-D), tile_dim2=0
  g1[5] = CHUNK_FLTS;                            // tensor_dim0_stride lo
  g1[6] = 0;
  g1[7] = 0;
  v4i gz4 = {0, 0, 0, 0};
  v8i gz8 = {0, 0, 0, 0, 0, 0, 0, 0};
  __builtin_amdgcn_tensor_load_to_lds(g0, g1, gz4, gz4, gz8, 0);
}

// ---------------------------------------------------------------------------
// Kernel 4: tiled WMMA renderer.
//  - block = 256 threads = 8 waves; each wave owns a 16x1 pixel strip
//  - sorted gaussians streamed through LDS via TENSOR_LOAD_TO_LDS (TDM),
//    double-buffered: wave 0 issues DMA for chunk c+1 while all waves
//    composite chunk c, then retires it with s_wait_tensorcnt before the
//    block barrier.
//  - per 16-gaussian subchunk: maha(16 gauss x 16 pix) via 2x
//    v_wmma_f32_16x16x4_f32 ; D = A(coeffs 16x8) x B(pixel monomials 8x16)
//  - D layout (wave32): lane L -> pixel N=L&15, gaussians M = r + 8*(L>>4)
//    -> in-lane sequential scan over 8 gaussians, one shuffle stitch per sub
// ---------------------------------------------------------------------------
__global__ void __launch_bounds__(256)
gs_render(const float* __restrict__ recS, float* __restrict__ out) {
  __shared__ __align__(16) float lds[2][CHUNK_FLTS];

  const int tid  = threadIdx.x;
  const int wave = tid >> 5;
  const int lane = tid & 31;
  const int half = lane >> 4;          // 0: gaussians 0-7, 1: gaussians 8-15
  const int pcol = lane & 15;          // pixel column within strip (D column N)

  const int strip = blockIdx.x * 8 + wave;   // 1024 strips of 16x1 pixels
  const int py  = strip >> 3;
  const int px0 = (strip & 7) << 4;

  // Pixel monomials f = [x^2, xy, y^2, x, y, 1, 0, 0]  (B matrix, per-lane slice)
  const float fx = (float)(px0 + pcol) + 0.5f;
  const float fy = (float)py + 0.5f;
  const float f[8] = { fx*fx, fx*fy, fy*fy, fx, fy, 1.0f, 0.0f, 0.0f };
  const v2f b1 = { f[2*half + 0], f[2*half + 1] };        // K = 0..3 slice
  const v2f b2 = { f[4 + 2*half + 0], f[4 + 2*half + 1] };// K = 4..7 slice

  const unsigned ldsAddr0 = (unsigned)(uintptr_t)&lds[0][0];
  const unsigned ldsAddr1 = (unsigned)(uintptr_t)&lds[1][0];

  float pixR = 0.0f, pixG = 0.0f, pixB = 0.0f;
  float Tg = 1.0f;                     // global transmittance for this pixel

  // stage chunk 0 via TDM (wave 0 issues; everyone waits at the barrier)
  if (wave == 0) {
    tdm_load_chunk(recS, ldsAddr0);
    __builtin_amdgcn_s_wait_tensorcnt(0);
  }
  __syncthreads();

  for (int c = 0; c < NCHUNK; ++c) {
    const int buf = c & 1;

    // overlap: DMA next chunk into the other buffer while we composite
    if (wave == 0 && c + 1 < NCHUNK)
      tdm_load_chunk(recS + (c + 1)*CHUNK_FLTS, buf ? ldsAddr0 : ldsAddr1);
    // warm L2 two chunks ahead (gfx1250 global_prefetch_b8)
    if (c + 2 < NCHUNK)
      __builtin_prefetch(recS + (c + 2)*CHUNK_FLTS + tid*6, 0, 1);

    #pragma unroll
    for (int sub = 0; sub < CHUNK/16; ++sub) {
      // A matrix slice: row M = lane&15, K slice per half
      const float* ar = &lds[buf][(sub*16 + pcol) * REC];
      const v2f a1 = { ar[2*half + 0], ar[2*half + 1] };
      const v2f a2 = { ar[4 + 2*half + 0], ar[4 + 2*half + 1] };

      v8f acc = {0.f,0.f,0.f,0.f,0.f,0.f,0.f,0.f};
      acc = __builtin_amdgcn_wmma_f32_16x16x4_f32(false, a1, false, b1,
                                                  (short)0, acc, false, false);
      acc = __builtin_amdgcn_wmma_f32_16x16x4_f32(false, a2, false, b2,
                                                  (short)0, acc, false, false);

      // local front-to-back scan over this lane's 8 gaussians (depth order)
      float localT = 1.0f, cr = 0.0f, cg = 0.0f, cb = 0.0f;
      #pragma unroll
      for (int r = 0; r < 8; ++r) {
        const int gl = sub*16 + half*8 + r;                      // index in chunk
        const float4 og = *(const float4*)&lds[buf][gl*REC + 8]; // opac, r, g, b
        float m  = fminf(acc[r], 20.0f);
        float al = og.x * __expf(-0.5f * m);
        float w  = localT * al;
        cr += w * og.y; cg += w * og.z; cb += w * og.w;
        localT *= (1.0f - al);
      }

      // stitch low half (gauss 0-7) then high half (8-15) in depth order
      float TLo  = __shfl(localT, pcol, 32);
      float crLo = __shfl(cr, pcol, 32);
      float cgLo = __shfl(cg, pcol, 32);
      float cbLo = __shfl(cb, pcol, 32);
      float THi  = __shfl(localT, pcol + 16, 32);
      float crHi = __shfl(cr, pcol + 16, 32);
      float cgHi = __shfl(cg, pcol + 16, 32);
      float cbHi = __shfl(cb, pcol + 16, 32);

      pixR += Tg * (crLo + TLo * crHi);
      pixG += Tg * (cgLo + TLo * cgHi);
      pixB += Tg * (cbLo + TLo * cbHi);
      Tg   *= TLo * THi;
    }

    // retire the in-flight DMA, then publish the new buffer to all waves.
    // The barrier also guarantees every wave is done reading buf before the
    // next iteration's TDM overwrites it.
    if (wave == 0 && c + 1 < NCHUNK)
      __builtin_amdgcn_s_wait_tensorcnt(0);
    __syncthreads();
  }

  if (lane < 16) {
    const int pix = py * IMG_W + px0 + lane;
    const float bg = 1.0f;
    out[0*IMG_H*IMG_W + pix] = fminf(fmaxf(pixR + Tg*bg, 0.0f), 1.0f);
    out[1*IMG_H*IMG_W + pix] = fminf(fmaxf(pixG + Tg*bg, 0.0f), 1.0f);
    out[2*IMG_H*IMG_W + pix] = fminf(fmaxf(pixB + Tg*bg, 0.0f), 1.0f);
  }
}

// ---------------------------------------------------------------------------
extern "C" void kernel_launch(void* const* d_in, const int* in_sizes, int n_in,
                              void* d_out, int out_size, void* d_ws, size_t ws_size,
                              hipStream_t stream) {
  const float* off = (const float*)d_in[0];
  const float* scl = (const float*)d_in[1];
  const float* opa = (const float*)d_in[2];
  const float* rot = (const float*)d_in[3];
  const float* col = (const float*)d_in[4];
  const float* vm  = (const float*)d_in[5];
  const float* pm  = (const float*)d_in[6];
  float* out = (float*)d_out;

  float* rec  = (float*)d_ws;          // N_G * 12
  float* recS = rec  + N_G * REC;      // N_G * 12
  float* keys = recS + N_G * REC;      // N_G
  int*   idx  = (int*)(keys + N_G);    // N_G

  gs_prep<<<N_G/256, 256, 0, stream>>>(off, scl, opa, rot, col, vm, pm, rec, keys, idx);
  gs_sort<<<1, 256, 0, stream>>>(keys, idx);
  gs_gather<<<N_G/256, 256, 0, stream>>>(rec, idx, recS);
  gs_render<<<(IMG_H*IMG_W)/(16*8), 256, 0, stream>>>(recS, out);
}